// AdaptiveHierarchicalQuantizer_27358941675812
// MI455X (gfx1250) — compile-verified
//
#include <hip/hip_runtime.h>

typedef __attribute__((ext_vector_type(2))) float v2f;
typedef __attribute__((ext_vector_type(8))) float v8f;

#define TOK_PER_WG 16
#define NWAVES     8
#define HDIM       1024
#define LAYERS     4
#define DDIM       256
#define KCODES     1024
#define N_TOKENS   (8 * 4096)

// Flat float offsets of the concatenated output tuple:
// (quantized, indices, quant, avg_code_probs, x)
#define SZ_BIG     ((size_t)N_TOKENS * HDIM)          // 33,554,432
#define OFF_QUANTD ((size_t)0)
#define OFF_IDX    (SZ_BIG)
#define SZ_IDX     ((size_t)N_TOKENS * LAYERS)        // 131,072
#define OFF_QUANT  (OFF_IDX + SZ_IDX)
#define OFF_AVGP   (OFF_QUANT + SZ_BIG)
#define OFF_X      (OFF_AVGP + (size_t)N_TOKENS * KCODES)

__global__ __launch_bounds__(256)
void ahq_kernel(const float* __restrict__ x,
                const float* __restrict__ cb,
                const float* __restrict__ temp,
                float* __restrict__ out)
{
    extern __shared__ float lds[];
    float* x_tile = lds;                       // 16 x 256  (current layer slice)
    float* sim    = lds + TOK_PER_WG * DDIM;   // 16 x 1024

    const int tid   = threadIdx.x;
    const int wave  = tid >> 5;
    const int lane  = tid & 31;
    const int row16 = lane & 15;
    const int hi    = lane >> 4;               // selects K-pair per ISA 7.12.2

    const int tok0 = blockIdx.x * TOK_PER_WG;

    float* out_q  = out + OFF_QUANTD;
    int*   out_i  = (int*)(out + OFF_IDX);
    float* out_q2 = out + OFF_QUANT;
    float* out_ap = out + OFF_AVGP;
    float* out_x  = out + OFF_X;

    const float tau     = fmaxf(temp[0], 0.04f);
    const float inv_tau = 1.0f / tau;

    // ---- x passthrough output: straight global->global float4 copy ----
    {
        const float4* src = (const float4*)(x + (size_t)tok0 * HDIM);
        float4*       dst = (float4*)(out_x + (size_t)tok0 * HDIM);
        const int nvec = TOK_PER_WG * HDIM / 4;   // 4096
#pragma unroll
        for (int i = 0; i < nvec / 256; ++i)
            dst[tid + i * 256] = src[tid + i * 256];
    }

    // per-lane prob accumulators for this wave's two tokens (codes lane+32*i)
    float accp0[32], accp1[32];
#pragma unroll
    for (int i = 0; i < 32; ++i) { accp0[i] = 0.f; accp1[i] = 0.f; }

    for (int l = 0; l < LAYERS; ++l) {
        const float* cb_l = cb + (size_t)l * KCODES * DDIM;

        // ---- stage x[:, l*256:(l+1)*256] slice into LDS (16x256) ----
        {
            const int nvec = TOK_PER_WG * DDIM / 4;   // 1024 float4
#pragma unroll
            for (int i = 0; i < nvec / 256; ++i) {
                int v   = tid + i * 256;
                int r   = v >> 6;                     // / (256/4)
                int c4  = v & 63;
                ((float4*)x_tile)[v] =
                    *(const float4*)(x + (size_t)(tok0 + r) * HDIM + l * DDIM + c4 * 4);
            }
        }
        __syncthreads();

        // ---- sim GEMM: this wave owns codes [wave*128, wave*128+128) ----
        v8f acc[8];
#pragma unroll
        for (int nt = 0; nt < 8; ++nt) acc[nt] = (v8f){};

        const int    code_base = wave * 128;
        const float* a_row     = x_tile + row16 * DDIM + hi * 2;
        // per-lane B row base; nt stride = 16 rows * DDIM floats (immediate offsets)
        const float* b_base    = cb_l + (size_t)(code_base + row16) * DDIM + hi * 2;

        // Software-pipelined B fragments: two explicit register buffers so 8
        // loads stay in flight while 8 WMMAs run on the other buffer.
        v2f bbuf0[8], bbuf1[8];
#pragma unroll
        for (int nt = 0; nt < 8; ++nt)
            bbuf0[nt] = *(const v2f*)(b_base + nt * 16 * DDIM);

        for (int d = 0; d < DDIM; d += 8) {
            // prefetch k-step d+4
#pragma unroll
            for (int nt = 0; nt < 8; ++nt)
                bbuf1[nt] = *(const v2f*)(b_base + nt * 16 * DDIM + d + 4);

            v2f a0 = *(const v2f*)(a_row + d);
#pragma unroll
            for (int nt = 0; nt < 8; ++nt)
                acc[nt] = __builtin_amdgcn_wmma_f32_16x16x4_f32(
                    false, a0, false, bbuf0[nt], (short)0, acc[nt], false, false);

            // prefetch k-step d+8 (skip past end on last iteration)
            if (d + 8 < DDIM) {
#pragma unroll
                for (int nt = 0; nt < 8; ++nt)
                    bbuf0[nt] = *(const v2f*)(b_base + nt * 16 * DDIM + d + 8);
            }

            v2f a1 = *(const v2f*)(a_row + d + 4);
#pragma unroll
            for (int nt = 0; nt < 8; ++nt)
                acc[nt] = __builtin_amdgcn_wmma_f32_16x16x4_f32(
                    false, a1, false, bbuf1[nt], (short)0, acc[nt], false, false);
        }

        // scatter scaled sim tiles to LDS (C layout: VGPR v -> M=v+hi*8, N=row16)
#pragma unroll
        for (int nt = 0; nt < 8; ++nt) {
#pragma unroll
            for (int v = 0; v < 8; ++v) {
                int m = v + hi * 8;
                sim[m * KCODES + code_base + nt * 16 + row16] = acc[nt][v] * inv_tau;
            }
        }
        __syncthreads();

        // ---- softmax + argmax: wave handles tokens 2*wave, 2*wave+1 ----
#pragma unroll
        for (int tt = 0; tt < 2; ++tt) {
            const int trow  = wave * 2 + tt;
            const int token = tok0 + trow;

            float vbuf[32];
#pragma unroll
            for (int i = 0; i < 32; ++i)
                vbuf[i] = sim[trow * KCODES + lane + 32 * i];

            // first-occurrence argmax (code index = lane + 32*i)
            float mval = vbuf[0];
            int   midx = lane;
#pragma unroll
            for (int i = 1; i < 32; ++i) {
                if (vbuf[i] > mval) { mval = vbuf[i]; midx = lane + 32 * i; }
            }
#pragma unroll
            for (int off = 16; off >= 1; off >>= 1) {
                float ov = __shfl_xor(mval, off, 32);
                int   oi = __shfl_xor(midx, off, 32);
                if (ov > mval || (ov == mval && oi < midx)) { mval = ov; midx = oi; }
            }

            // exp / sum / normalize, accumulate layer-mean probs
            float ssum = 0.f;
#pragma unroll
            for (int i = 0; i < 32; ++i) { vbuf[i] = __expf(vbuf[i] - mval); ssum += vbuf[i]; }
#pragma unroll
            for (int off = 16; off >= 1; off >>= 1) ssum += __shfl_xor(ssum, off, 32);
            const float inv_sum = 1.0f / ssum;

#pragma unroll
            for (int i = 0; i < 32; ++i) {
                float p = vbuf[i] * inv_sum;
                if (tt == 0) accp0[i] += p; else accp1[i] += p;
            }

            if (lane == 0)
                out_i[(size_t)token * LAYERS + l] = midx;

            // hard_q gather: straight-through forward value == codebook row.
            // Write to both 'quantized' and 'quant' regions (identical flat layout).
            const float4* crow = (const float4*)(cb_l + (size_t)midx * DDIM);
            float4* q0 = (float4*)(out_q  + (size_t)token * HDIM + l * DDIM);
            float4* q1 = (float4*)(out_q2 + (size_t)token * HDIM + l * DDIM);
#pragma unroll
            for (int j = 0; j < 2; ++j) {
                float4 v = crow[lane + 32 * j];
                q0[lane + 32 * j] = v;
                q1[lane + 32 * j] = v;
            }
        }
        __syncthreads();
    }

    // ---- avg_code_probs = mean over 4 layers ----
    {
        const int trow = wave * 2;
        float* ap0 = out_ap + (size_t)(tok0 + trow)     * KCODES;
        float* ap1 = out_ap + (size_t)(tok0 + trow + 1) * KCODES;
#pragma unroll
        for (int i = 0; i < 32; ++i) {
            ap0[lane + 32 * i] = accp0[i] * 0.25f;
            ap1[lane + 32 * i] = accp1[i] * 0.25f;
        }
    }
}

extern "C" void kernel_launch(void* const* d_in, const int* in_sizes, int n_in,
                              void* d_out, int out_size, void* d_ws, size_t ws_size,
                              hipStream_t stream) {
    const float* x    = (const float*)d_in[0];   // (8, 4096, 1024) f32
    const float* cb   = (const float*)d_in[1];   // (4, 1024, 256)  f32
    const float* temp = (const float*)d_in[2];   // (1,)            f32
    float* out = (float*)d_out;

    const int    nblocks   = N_TOKENS / TOK_PER_WG;                        // 2048
    const size_t lds_bytes = (TOK_PER_WG * DDIM + TOK_PER_WG * KCODES)
                             * sizeof(float);                              // 80 KB

    ahq_kernel<<<dim3(nblocks), dim3(256), lds_bytes, stream>>>(x, cb, temp, out);
}